// SHGAT_14680198218451
// MI455X (gfx1250) — compile-verified
//
#include <hip/hip_runtime.h>
#include <hip/hip_bf16.h>
#include <math.h>

typedef __attribute__((ext_vector_type(2))) float v2f;
typedef __attribute__((ext_vector_type(8))) float v8f;

#define DIM   64
#define NNBR  64
#define TOPK  8
#define NREL  8
#define NEGV  (-9000000000000000.0f)

// CDNA5 async global->LDS copy (ASYNCcnt-tracked, no VGPR data path).
__device__ __forceinline__ void async_ld_b128(unsigned lds_off, const void* gaddr)
{
    asm volatile("global_load_async_to_lds_b128 %0, %1, off"
                 :: "v"(lds_off), "v"(gaddr) : "memory");
}
__device__ __forceinline__ void wait_async0()
{
    asm volatile("s_wait_asynccnt 0x0" ::: "memory");
}

// ---------------------------------------------------------------------------
// Kernel 1: user embedding = mean of clicked-item embeddings (scaled by 1/num)
// ---------------------------------------------------------------------------
__global__ __launch_bounds__(64) void shgat_user_emb(
    const int* __restrict__ user_indices,
    const float* __restrict__ entity_emb,
    const int* __restrict__ items_clicked,
    const float* __restrict__ num_clicked,
    float* __restrict__ user_emb, int hist)
{
    int b = blockIdx.x;
    int d = threadIdx.x;
    int u = user_indices[b];
    const int* hrow = items_clicked + (size_t)u * hist;
    float acc = 0.0f;
    for (int h = 0; h < hist; ++h) {
        int e = hrow[h];
        acc += entity_emb[(size_t)e * DIM + d];
    }
    user_emb[(size_t)b * DIM + d] = acc * (1.0f / num_clicked[u]);
}

// ---------------------------------------------------------------------------
// Kernel 2: hop expansion. One 128-thread block (4 waves) per parent node.
//   Y[j,:] = tanh(W_rel[r_j] . emb[e_j])  for 64 neighbors j  (WMMA f32)
//   s_j = mean(user_emb * Y[j])  -> masked softmax(64) -> top8 -> softmax(8)
//
// WMMA trick: relations are row-disjoint, so Y = sum_r (A_r x W_r^T) where
// A_r zeroes rows whose relation != r. Exact in f32 (zeros contribute 0).
// Data movement: neighbor embeddings + W_rel staged via async global->LDS;
// W_rel double-buffered so the prefetch of W_{r+1} overlaps WMMA on W_r.
// ---------------------------------------------------------------------------
__global__ __launch_bounds__(128) void shgat_expand(
    const float* __restrict__ entity_emb,
    const float* __restrict__ W_rel,
    const int* __restrict__ adj_entity,
    const int* __restrict__ adj_relation,
    const float* __restrict__ labels_table,
    const int* __restrict__ user_indices,
    const int* __restrict__ item_indices,
    const float* __restrict__ user_emb,
    const int* __restrict__ parents,   // entity ids, one per block
    int parents_per_b,
    int n_ent_p1,
    float* __restrict__ s_out,         // [npar*8]
    float* __restrict__ y_out,         // [npar*8*64]
    float* __restrict__ lab_out,       // [npar*8]
    float* __restrict__ reset_out,     // [npar*8] or nullptr
    int*   __restrict__ esel_out)      // [npar*8] or nullptr
{
    __shared__ float embS[NNBR * DIM];       // neighbor embeddings (row major)
    __shared__ float wS[2][DIM * DIM];       // double-buffered W_r[e][d]
    __shared__ float yS[NNBR * DIM];         // transformed neighbors
    __shared__ float uS[DIM];
    __shared__ float sS[NNBR];
    __shared__ int   relS[NNBR];
    __shared__ int   entS[NNBR];
    __shared__ int   idsS[TOPK];
    __shared__ float stopS[TOPK];

    const int bp  = blockIdx.x;
    const int b   = bp / parents_per_b;
    const int tid = threadIdx.x;
    const int parent = parents[bp];

    if (tid < NNBR) {
        entS[tid] = adj_entity[(size_t)parent * NNBR + tid];
        relS[tid] = adj_relation[(size_t)parent * NNBR + tid];
    }
    if (tid < DIM) uS[tid] = user_emb[(size_t)b * DIM + tid];
    __syncthreads();

    // async-stage the 64 gathered neighbor embedding rows (16 KB)
    for (int idx = tid * 4; idx < NNBR * DIM; idx += 128 * 4) {
        int j = idx >> 6, d = idx & 63;
        async_ld_b128((unsigned)(uintptr_t)&embS[idx],
                      entity_emb + (size_t)entS[j] * DIM + d);
    }
    // async-stage W_rel[0] into buffer 0 (16 KB)
    for (int idx = tid * 4; idx < DIM * DIM; idx += 128 * 4)
        async_ld_b128((unsigned)(uintptr_t)&wS[0][idx], W_rel + idx);

    const int wave  = tid >> 5;      // row tile (16 neighbors) handled by wave
    const int lane  = tid & 31;
    const int mrow  = lane & 15;
    const int khalf = lane >> 4;     // lanes 16-31 hold K+2,K+3
    const int jrow  = wave * 16 + mrow;

    v8f c0 = {}, c1 = {}, c2 = {}, c3 = {};

    for (int r = 0; r < NREL; ++r) {
        wait_async0();           // prefetched buffer (and embS on r==0) landed
        __syncthreads();         // all waves done reading buffer being reused

        if (r + 1 < NREL) {      // prefetch W_{r+1} while computing on W_r
            const float* wnext = W_rel + (size_t)(r + 1) * DIM * DIM;
            float* dst = wS[(r + 1) & 1];
            for (int idx = tid * 4; idx < DIM * DIM; idx += 128 * 4)
                async_ld_b128((unsigned)(uintptr_t)&dst[idx], wnext + idx);
        }
        const float* __restrict__ w = wS[r & 1];

        const bool on = (relS[jrow] == r);
        for (int kk = 0; kk < 16; ++kk) {
            const int k0 = kk * 4 + khalf * 2;
            v2f a;
            a.x = on ? embS[jrow * DIM + k0]     : 0.0f;
            a.y = on ? embS[jrow * DIM + k0 + 1] : 0.0f;
            {   // N-tile 0
                v2f bf; int n = 0 * 16 + mrow;
                bf.x = w[n * DIM + k0]; bf.y = w[n * DIM + k0 + 1];
                c0 = __builtin_amdgcn_wmma_f32_16x16x4_f32(false, a, false, bf,
                                                           (short)0, c0, false, false);
            }
            {   // N-tile 1
                v2f bf; int n = 1 * 16 + mrow;
                bf.x = w[n * DIM + k0]; bf.y = w[n * DIM + k0 + 1];
                c1 = __builtin_amdgcn_wmma_f32_16x16x4_f32(false, a, false, bf,
                                                           (short)0, c1, false, false);
            }
            {   // N-tile 2
                v2f bf; int n = 2 * 16 + mrow;
                bf.x = w[n * DIM + k0]; bf.y = w[n * DIM + k0 + 1];
                c2 = __builtin_amdgcn_wmma_f32_16x16x4_f32(false, a, false, bf,
                                                           (short)0, c2, false, false);
            }
            {   // N-tile 3
                v2f bf; int n = 3 * 16 + mrow;
                bf.x = w[n * DIM + k0]; bf.y = w[n * DIM + k0 + 1];
                c3 = __builtin_amdgcn_wmma_f32_16x16x4_f32(false, a, false, bf,
                                                           (short)0, c3, false, false);
            }
        }
    }
    __syncthreads();

    // C layout: lanes 0-15 -> rows M=v, lanes 16-31 -> rows M=v+8, N = tile*16+mrow
    #pragma unroll
    for (int v = 0; v < 8; ++v) {
        int row = wave * 16 + ((lane < 16) ? v : v + 8);
        yS[row * DIM +  0 + mrow] = tanhf(c0[v]);
        yS[row * DIM + 16 + mrow] = tanhf(c1[v]);
        yS[row * DIM + 32 + mrow] = tanhf(c2[v]);
        yS[row * DIM + 48 + mrow] = tanhf(c3[v]);
    }
    __syncthreads();

    // attention scores: s_j = mean_d(user_emb_d * Y[j,d])
    if (tid < NNBR) {
        float acc = 0.0f;
        for (int d = 0; d < DIM; ++d) acc += uS[d] * yS[tid * DIM + d];
        sS[tid] = acc * (1.0f / DIM);
    }
    __syncthreads();

    if (tid == 0) {
        // masked softmax over 64 neighbors
        float tmp[NNBR];
        float mx = -INFINITY;
        for (int j = 0; j < NNBR; ++j) {
            float v = sS[j];
            v = (v == 0.0f) ? NEGV : v;
            tmp[j] = v;
            mx = fmaxf(mx, v);
        }
        float sum = 0.0f;
        for (int j = 0; j < NNBR; ++j) { tmp[j] = expf(tmp[j] - mx); sum += tmp[j]; }
        float inv = 1.0f / sum;
        for (int j = 0; j < NNBR; ++j) sS[j] = tmp[j] * inv;

        // top-8 (stable: strict >, lowest index wins ties, descending order)
        bool used[NNBR];
        for (int j = 0; j < NNBR; ++j) used[j] = false;
        for (int k = 0; k < TOPK; ++k) {
            int best = 0; float bv = -INFINITY;
            for (int j = 0; j < NNBR; ++j)
                if (!used[j] && sS[j] > bv) { bv = sS[j]; best = j; }
            used[best] = true;
            idsS[k] = best;
            stopS[k] = bv;
        }
        // second masked softmax over top-8
        float t2[TOPK];
        float mx2 = -INFINITY;
        for (int k = 0; k < TOPK; ++k) {
            float v = stopS[k];
            v = (v == 0.0f) ? NEGV : v;
            t2[k] = v;
            mx2 = fmaxf(mx2, v);
        }
        float sum2 = 0.0f;
        for (int k = 0; k < TOPK; ++k) { t2[k] = expf(t2[k] - mx2); sum2 += t2[k]; }
        float inv2 = 1.0f / sum2;
        for (int k = 0; k < TOPK; ++k) stopS[k] = t2[k] * inv2;
    }
    __syncthreads();

    // copy selected rows + scores
    for (int idx = tid; idx < TOPK * DIM; idx += 128) {
        int k = idx >> 6, d = idx & 63;
        y_out[((size_t)bp * TOPK + k) * DIM + d] = yS[idsS[k] * DIM + d];
    }
    if (tid < TOPK) {
        s_out[(size_t)bp * TOPK + tid] = stopS[tid];
        int e = entS[idsS[tid]];
        if (esel_out) esel_out[(size_t)bp * TOPK + tid] = e;
        int u    = user_indices[b];
        int seed = item_indices[b];
        float lr = labels_table[(size_t)u * n_ent_p1 + e];
        bool mask = (e != seed);                  // holdout mask (same user)
        lab_out[(size_t)bp * TOPK + tid] = mask ? lr : 0.5f;
        if (reset_out)
            reset_out[(size_t)bp * TOPK + tid] = (mask && (lr != 0.5f)) ? 1.0f : 0.0f;
    }
}

// ---------------------------------------------------------------------------
// Kernel 3: aggregation + final scores. One 64-thread block per batch element.
// ---------------------------------------------------------------------------
__global__ __launch_bounds__(64) void shgat_finalize(
    const float* __restrict__ entity_emb,
    const int* __restrict__ item_indices,
    const float* __restrict__ user_emb,
    const float* __restrict__ s1,      // [bs*8]
    const float* __restrict__ y1,      // [bs*8*64]
    const float* __restrict__ lab1,    // [bs*8]
    const float* __restrict__ reset1,  // [bs*8]
    const float* __restrict__ s2,      // [bs*8*8]
    const float* __restrict__ y2,      // [bs*8*8*64]
    const float* __restrict__ lab2,    // [bs*8*8]
    float* __restrict__ out,           // [2*bs]: scores then pred_labels
    int bs)
{
    __shared__ float s1S[TOPK];
    __shared__ float red[DIM];

    int b = blockIdx.x, tid = threadIdx.x;
    if (tid < TOPK) s1S[tid] = s1[(size_t)b * TOPK + tid];
    __syncthreads();

    const int e = tid;
    float seed_e = entity_emb[(size_t)item_indices[b] * DIM + e];
    float acc0 = 0.0f;  // sum_p s1*y1      -> emb0_new
    float accF = 0.0f;  // sum_p s1*emb1new -> final
    for (int p = 0; p < TOPK; ++p) {
        size_t bp  = (size_t)b * TOPK + p;
        float y1e  = y1[bp * DIM + e];
        acc0 += s1S[p] * y1e;
        float a = 0.0f;
        for (int k = 0; k < TOPK; ++k)
            a += s2[bp * TOPK + k] * y2[(bp * TOPK + k) * DIM + e];
        float e1 = a * (1.0f / TOPK) + y1e;
        e1 = fmaxf(e1, 0.0f);                       // relu (iter 0, hop 1)
        accF += s1S[p] * e1;
    }
    float emb0 = fmaxf(acc0 * (1.0f / TOPK) + seed_e, 0.0f);  // relu (iter 0, hop 0)
    float fin  = tanhf(accF * (1.0f / TOPK) + emb0);          // tanh (iter 1)

    red[tid] = user_emb[(size_t)b * DIM + e] * fin;
    __syncthreads();
    for (int off = 32; off > 0; off >>= 1) {
        if (tid < off) red[tid] += red[tid + off];
        __syncthreads();
    }
    if (tid == 0)
        out[b] = 1.0f / (1.0f + expf(-red[0]));

    if (tid == 0) {
        float labF = 0.0f;
        for (int p = 0; p < TOPK; ++p) {
            size_t bp = (size_t)b * TOPK + p;
            float l1n;
            if (reset1[bp] != 0.0f) l1n = lab1[bp];
            else {
                float a = 0.0f;
                for (int k = 0; k < TOPK; ++k)
                    a += s2[bp * TOPK + k] * lab2[bp * TOPK + k];
                l1n = a;
            }
            labF += s1S[p] * l1n;
        }
        out[bs + b] = 1.0f / (1.0f + expf(-(labF - 0.5f)));
    }
}

// ---------------------------------------------------------------------------
// launch
// ---------------------------------------------------------------------------
extern "C" void kernel_launch(void* const* d_in, const int* in_sizes, int n_in,
                              void* d_out, int out_size, void* d_ws, size_t ws_size,
                              hipStream_t stream)
{
    const int*   user_indices = (const int*)  d_in[0];
    const int*   item_indices = (const int*)  d_in[1];
    const float* entity_emb   = (const float*)d_in[2];
    const float* W_rel        = (const float*)d_in[3];
    const int*   adj_entity   = (const int*)  d_in[4];
    const int*   adj_relation = (const int*)  d_in[5];
    const int*   items_clicked= (const int*)  d_in[6];
    const float* num_clicked  = (const float*)d_in[7];
    const float* labels_table = (const float*)d_in[8];

    const int bs       = in_sizes[0];
    const int n_ent_p1 = in_sizes[2] / DIM;
    const int hist     = in_sizes[6] / bs;

    float* ws = (float*)d_ws;
    size_t o = 0;
    float* user_emb = ws + o; o += (size_t)bs * DIM;
    float* s1    = ws + o; o += (size_t)bs * TOPK;
    float* y1    = ws + o; o += (size_t)bs * TOPK * DIM;
    float* lab1  = ws + o; o += (size_t)bs * TOPK;
    float* rst1  = ws + o; o += (size_t)bs * TOPK;
    int*   esel1 = (int*)(ws + o); o += (size_t)bs * TOPK;
    float* s2    = ws + o; o += (size_t)bs * TOPK * TOPK;
    float* y2    = ws + o; o += (size_t)bs * TOPK * TOPK * DIM;
    float* lab2  = ws + o; o += (size_t)bs * TOPK * TOPK;

    shgat_user_emb<<<bs, 64, 0, stream>>>(user_indices, entity_emb, items_clicked,
                                          num_clicked, user_emb, hist);

    // level 1: parents = seed items (1 parent per batch element)
    shgat_expand<<<bs, 128, 0, stream>>>(entity_emb, W_rel, adj_entity, adj_relation,
                                         labels_table, user_indices, item_indices,
                                         user_emb, item_indices, /*parents_per_b=*/1,
                                         n_ent_p1, s1, y1, lab1, rst1, esel1);

    // level 2: parents = selected hop-1 entities (8 per batch element)
    shgat_expand<<<bs * TOPK, 128, 0, stream>>>(entity_emb, W_rel, adj_entity,
                                                adj_relation, labels_table,
                                                user_indices, item_indices,
                                                user_emb, esel1, /*parents_per_b=*/TOPK,
                                                n_ent_p1, s2, y2, lab2,
                                                nullptr, nullptr);

    shgat_finalize<<<bs, 64, 0, stream>>>(entity_emb, item_indices, user_emb,
                                          s1, y1, lab1, rst1, s2, y2, lab2,
                                          (float*)d_out, bs);
}